// DecoderLayerForSherpa_61229053772469
// MI455X (gfx1250) — compile-verified
//
#include <hip/hip_runtime.h>
#include <math.h>

typedef __attribute__((ext_vector_type(2))) float v2f;
typedef __attribute__((ext_vector_type(8))) float v8f;
typedef __attribute__((ext_vector_type(4))) int   v4i;

#define EMBED  1024
#define FFNDIM 4096
#define BSZ    32
#define HEADS  16
#define HD     64
#define SELFC  448
#define CROSSS 1500

#if defined(__has_builtin)
#if __has_builtin(__builtin_amdgcn_global_load_async_to_lds_b128) && \
    __has_builtin(__builtin_amdgcn_global_store_async_from_lds_b128) && \
    __has_builtin(__builtin_amdgcn_s_wait_asynccnt)
#define USE_ASYNC_LDS 1
#endif
#endif

#if defined(USE_ASYNC_LDS)
typedef __attribute__((address_space(1))) v4i* gptr_b128;
typedef __attribute__((address_space(3))) v4i* lptr_b128;

__device__ __forceinline__ void async_g2l_b128(const float* g, float* l) {
  __builtin_amdgcn_global_load_async_to_lds_b128((gptr_b128)g, (lptr_b128)l, 0, 0);
}
__device__ __forceinline__ void async_l2g_b128(float* g, const float* l) {
  __builtin_amdgcn_global_store_async_from_lds_b128((gptr_b128)g, (lptr_b128)l, 0, 0);
}
#endif

// ---------------------------------------------------------------------------
// LayerNorm over 32 rows of 1024 (one block per row)
// ---------------------------------------------------------------------------
__global__ __launch_bounds__(256) void ln_rows(const float* __restrict__ x,
                                               const float* __restrict__ g,
                                               const float* __restrict__ b,
                                               float* __restrict__ out) {
  __shared__ float rsum[8], rsq[8];
  const int row = blockIdx.x;
  const float* xr = x + (size_t)row * EMBED;
  float ls = 0.f, lq = 0.f;
  for (int i = threadIdx.x; i < EMBED; i += 256) {
    float v = xr[i]; ls += v; lq += v * v;
  }
  for (int off = 16; off > 0; off >>= 1) {
    ls += __shfl_xor(ls, off, 32);
    lq += __shfl_xor(lq, off, 32);
  }
  const int lane = threadIdx.x & 31, wv = threadIdx.x >> 5;
  if (lane == 0) { rsum[wv] = ls; rsq[wv] = lq; }
  __syncthreads();
  float ts = 0.f, tq = 0.f;
  for (int i = 0; i < 8; ++i) { ts += rsum[i]; tq += rsq[i]; }
  const float mean = ts * (1.0f / EMBED);
  const float var  = tq * (1.0f / EMBED) - mean * mean;
  const float inv  = rsqrtf(var + 1e-5f);
  for (int i = threadIdx.x; i < EMBED; i += 256) {
    out[(size_t)row * EMBED + i] = (xr[i] - mean) * inv * g[i] + b[i];
  }
}

// ---------------------------------------------------------------------------
// GEMM: OUT[32,N] = A[32,K] @ W[N,K]^T + bias, optional epilogue + residual.
// WMMA f32 16x16x4, wave32, software-pipelined fragment loads.
// Block = 256 threads = 8 waves = 128 N-cols. grid.x = N/128 exactly.
// mode: 0 = none, 1 = *0.125 (attention q scale), 2 = exact-erf GELU
// ---------------------------------------------------------------------------
__global__ __launch_bounds__(256) void gemm32(const float* __restrict__ A,
                                              const float* __restrict__ W,
                                              const float* __restrict__ bias,
                                              const float* __restrict__ resid,
                                              float* __restrict__ out,
                                              int K, int N, int mode) {
  const int lane  = threadIdx.x & 31;
  const int wave  = threadIdx.x >> 5;
  const int n0    = blockIdx.x * 128 + wave * 16;
  const int mrow  = lane & 15;               // M (A frag) / N (B frag) index
  const int khalf = (lane >> 4) << 1;        // 0 for lanes 0-15, 2 for 16-31
  const int ncol  = n0 + mrow;

  const float* a0p = A + (size_t)mrow * K + khalf;         // rows 0..15
  const float* a1p = A + (size_t)(mrow + 16) * K + khalf;  // rows 16..31
  const float* wp  = W + (size_t)ncol * K + khalf;         // B = W^T column

  v8f acc0 = {}, acc1 = {};
  // software pipeline: current fragments in registers, preload next chunk
  float2 ca0 = *(const float2*)(a0p);
  float2 ca1 = *(const float2*)(a1p);
  float2 cb  = *(const float2*)(wp);
#pragma unroll 4
  for (int k0 = 0; k0 < K - 4; k0 += 4) {
    const float2 na0 = *(const float2*)(a0p + k0 + 4);
    const float2 na1 = *(const float2*)(a1p + k0 + 4);
    const float2 nb  = *(const float2*)(wp + k0 + 4);
    v2f a0 = {ca0.x, ca0.y};
    v2f a1 = {ca1.x, ca1.y};
    v2f bb = {cb.x, cb.y};
    acc0 = __builtin_amdgcn_wmma_f32_16x16x4_f32(false, a0, false, bb,
                                                 (short)0, acc0, false, false);
    acc1 = __builtin_amdgcn_wmma_f32_16x16x4_f32(false, a1, false, bb,
                                                 (short)0, acc1, false, false);
    ca0 = na0; ca1 = na1; cb = nb;
  }
  {
    v2f a0 = {ca0.x, ca0.y};
    v2f a1 = {ca1.x, ca1.y};
    v2f bb = {cb.x, cb.y};
    acc0 = __builtin_amdgcn_wmma_f32_16x16x4_f32(false, a0, false, bb,
                                                 (short)0, acc0, false, false);
    acc1 = __builtin_amdgcn_wmma_f32_16x16x4_f32(false, a1, false, bb,
                                                 (short)0, acc1, false, false);
  }

  const float bv = bias[ncol];
  const int rb = (lane >> 4) << 3;  // C/D layout: lanes 16-31 hold M = i+8
#pragma unroll
  for (int i = 0; i < 8; ++i) {
    const int r0 = rb + i;
    const int r1 = rb + i + 16;
    float v0 = acc0[i] + bv;
    float v1 = acc1[i] + bv;
    if (mode == 1) {
      v0 *= 0.125f; v1 *= 0.125f;
    } else if (mode == 2) {
      v0 = 0.5f * v0 * (1.0f + erff(v0 * 0.70710678118654752f));
      v1 = 0.5f * v1 * (1.0f + erff(v1 * 0.70710678118654752f));
    }
    if (resid) {
      v0 += resid[(size_t)r0 * N + ncol];
      v1 += resid[(size_t)r1 * N + ncol];
    }
    out[(size_t)r0 * N + ncol] = v0;
    out[(size_t)r1 * N + ncol] = v1;
  }
}

// ---------------------------------------------------------------------------
// KV-cache bulk copy (rows 0..446 of each batch) via CDNA5 async Global<->LDS
// DMA when available (ASYNCcnt path), else float4 VGPR streaming.
// Each lane moves 64B of K and 64B of V per iteration; LDS slices are
// wave-local so no workgroup barriers are required.
// ---------------------------------------------------------------------------
__global__ __launch_bounds__(256) void cache_bulk(const float* __restrict__ kin,
                                                  const float* __restrict__ vin,
                                                  float* __restrict__ kout,
                                                  float* __restrict__ vout) {
  const size_t per_batch_units = (size_t)(SELFC - 1) * EMBED / 16;  // 64B units
  const size_t units = (size_t)BSZ * per_batch_units;
  const size_t stride = (size_t)gridDim.x * blockDim.x;
  size_t u = (size_t)blockIdx.x * blockDim.x + threadIdx.x;
#if defined(USE_ASYNC_LDS)
  __shared__ __align__(16) float ldsK[256 * 16];
  __shared__ __align__(16) float ldsV[256 * 16];
  float* lk = &ldsK[threadIdx.x * 16];
  float* lv = &ldsV[threadIdx.x * 16];
  for (; u < units; u += stride) {
    const size_t b = u / per_batch_units;
    const size_t off = (u - b * per_batch_units) * 16;            // floats
    const size_t g = b * (size_t)SELFC * EMBED + off;
#pragma unroll
    for (int j = 0; j < 4; ++j) {
      async_g2l_b128(kin + g + 4 * j, lk + 4 * j);
      async_g2l_b128(vin + g + 4 * j, lv + 4 * j);
    }
    __builtin_amdgcn_s_wait_asynccnt(0);   // LDS slice now holds the tile
#pragma unroll
    for (int j = 0; j < 4; ++j) {
      async_l2g_b128(kout + g + 4 * j, lk + 4 * j);
      async_l2g_b128(vout + g + 4 * j, lv + 4 * j);
    }
    __builtin_amdgcn_s_wait_asynccnt(0);   // safe to reuse LDS slice
  }
#else
  for (; u < units; u += stride) {
    const size_t b = u / per_batch_units;
    const size_t off = (u - b * per_batch_units) * 16;            // floats
    const size_t g = b * (size_t)SELFC * EMBED + off;
#pragma unroll
    for (int j = 0; j < 4; ++j) {
      *(float4*)(kout + g + 4 * j) = *(const float4*)(kin + g + 4 * j);
      *(float4*)(vout + g + 4 * j) = *(const float4*)(vin + g + 4 * j);
    }
  }
#endif
}

// ---------------------------------------------------------------------------
// Write the freshly projected k/v into the last sequence row of each batch.
// grid = 32 blocks x 256 threads : one float4 of k and v per thread.
// ---------------------------------------------------------------------------
__global__ __launch_bounds__(256) void cache_lastrow(const float* __restrict__ knew,
                                                     const float* __restrict__ vnew,
                                                     float* __restrict__ kout,
                                                     float* __restrict__ vout) {
  const int i = blockIdx.x * 256 + threadIdx.x;   // float4 index, 8192 total
  const int b = i >> 8;                           // 256 float4 per 1024-f row
  const int e = (i & 255) * 4;
  const size_t dst = ((size_t)b * SELFC + (SELFC - 1)) * EMBED + e;
  *(float4*)(kout + dst) = *(const float4*)(knew + (size_t)b * EMBED + e);
  *(float4*)(vout + dst) = *(const float4*)(vnew + (size_t)b * EMBED + e);
}

// ---------------------------------------------------------------------------
// T=1 attention for one (batch, head) per block. q pre-scaled by 1/8.
// smem: [64 q][S scores][256 reduce/partials]
// ---------------------------------------------------------------------------
__global__ __launch_bounds__(256) void attn_t1(const float* __restrict__ q,
                                               const float* __restrict__ kc,
                                               const float* __restrict__ vc,
                                               const float* __restrict__ mask,
                                               float* __restrict__ out, int S) {
  extern __shared__ float smem[];
  float* sq  = smem;            // 64
  float* sp  = smem + 64;       // S
  float* red = smem + 64 + S;   // 256
  const int b = blockIdx.x >> 4;
  const int h = blockIdx.x & 15;
  const int t = threadIdx.x;

  const float* qp = q + (size_t)b * EMBED + h * HD;
  if (t < HD) sq[t] = qp[t];
  __syncthreads();

  float lmax = -3.4e38f;
  for (int s = t; s < S; s += 256) {
    const float* kr = kc + ((size_t)b * S + s) * EMBED + h * HD;
    if (s + 256 < S)
      __builtin_prefetch(kc + ((size_t)b * S + s + 256) * EMBED + h * HD, 0, 0);
    float acc = 0.f;
#pragma unroll
    for (int d = 0; d < HD; d += 4) {
      const float4 kk = *(const float4*)(kr + d);
      acc += sq[d] * kk.x + sq[d + 1] * kk.y + sq[d + 2] * kk.z + sq[d + 3] * kk.w;
    }
    if (mask) acc += mask[s];
    sp[s] = acc;
    lmax = fmaxf(lmax, acc);
  }
  red[t] = lmax; __syncthreads();
  for (int off = 128; off > 0; off >>= 1) {
    if (t < off) red[t] = fmaxf(red[t], red[t + off]);
    __syncthreads();
  }
  const float mx = red[0]; __syncthreads();

  float lsum = 0.f;
  for (int s = t; s < S; s += 256) {
    const float e = expf(sp[s] - mx);
    sp[s] = e; lsum += e;
  }
  red[t] = lsum; __syncthreads();
  for (int off = 128; off > 0; off >>= 1) {
    if (t < off) red[t] += red[t + off];
    __syncthreads();
  }
  const float inv = 1.0f / red[0]; __syncthreads();

  const int d = t & 63, chunk = t >> 6;
  float acc = 0.f;
  for (int s = chunk; s < S; s += 4) {
    if (s + 4 < S)
      __builtin_prefetch(vc + ((size_t)b * S + s + 4) * EMBED + h * HD + d, 0, 0);
    acc += sp[s] * vc[((size_t)b * S + s) * EMBED + h * HD + d];
  }
  red[t] = acc; __syncthreads();
  if (t < 64) {
    const float tot = (red[t] + red[t + 64] + red[t + 128] + red[t + 192]) * inv;
    out[(size_t)b * EMBED + h * HD + d] = tot;
  }
}

// ---------------------------------------------------------------------------
extern "C" void kernel_launch(void* const* d_in, const int* in_sizes, int n_in,
                              void* d_out, int out_size, void* d_ws, size_t ws_size,
                              hipStream_t stream) {
  const float* x        = (const float*)d_in[0];
  const float* kcache   = (const float*)d_in[1];
  const float* vcache   = (const float*)d_in[2];
  const float* cross_k  = (const float*)d_in[3];
  const float* cross_v  = (const float*)d_in[4];
  const float* mask     = (const float*)d_in[5];
  const float* ln1_g    = (const float*)d_in[6];
  const float* ln1_b    = (const float*)d_in[7];
  const float* sa_qw    = (const float*)d_in[8];
  const float* sa_qb    = (const float*)d_in[9];
  const float* sa_kw    = (const float*)d_in[10];
  const float* sa_kb    = (const float*)d_in[11];
  const float* sa_vw    = (const float*)d_in[12];
  const float* sa_vb    = (const float*)d_in[13];
  const float* sa_ow    = (const float*)d_in[14];
  const float* sa_ob    = (const float*)d_in[15];
  const float* ln2_g    = (const float*)d_in[16];
  const float* ln2_b    = (const float*)d_in[17];
  const float* ca_qw    = (const float*)d_in[18];
  const float* ca_qb    = (const float*)d_in[19];
  const float* ca_ow    = (const float*)d_in[20];
  const float* ca_ob    = (const float*)d_in[21];
  const float* ln3_g    = (const float*)d_in[22];
  const float* ln3_b    = (const float*)d_in[23];
  const float* fc1_w    = (const float*)d_in[24];
  const float* fc1_b    = (const float*)d_in[25];
  const float* fc2_w    = (const float*)d_in[26];
  const float* fc2_b    = (const float*)d_in[27];

  float* ws   = (float*)d_ws;
  float* h    = ws;                 // [32,1024]
  float* qv   = ws + 32768;         // [32,1024]
  float* knew = ws + 65536;         // [32,1024]
  float* vnew = ws + 98304;         // [32,1024]
  float* att  = ws + 131072;        // [32,1024]
  float* x1   = ws + 163840;        // [32,1024]
  float* x2   = ws + 196608;        // [32,1024]
  float* mid  = ws + 229376;        // [32,4096]

  float* out_x = (float*)d_out;
  float* out_k = out_x + (size_t)BSZ * EMBED;
  float* out_v = out_k + (size_t)BSZ * SELFC * EMBED;

  const int smem_self  = (64 + SELFC  + 256) * sizeof(float);
  const int smem_cross = (64 + CROSSS + 256) * sizeof(float);

  // ---- self-attention ----
  ln_rows<<<BSZ, 256, 0, stream>>>(x, ln1_g, ln1_b, h);
  gemm32<<<EMBED / 128, 256, 0, stream>>>(h, sa_qw, sa_qb, nullptr, qv, EMBED, EMBED, 1);
  gemm32<<<EMBED / 128, 256, 0, stream>>>(h, sa_kw, sa_kb, nullptr, knew, EMBED, EMBED, 0);
  gemm32<<<EMBED / 128, 256, 0, stream>>>(h, sa_vw, sa_vb, nullptr, vnew, EMBED, EMBED, 0);
  cache_bulk<<<2048, 256, 0, stream>>>(kcache, vcache, out_k, out_v);
  cache_lastrow<<<BSZ, 256, 0, stream>>>(knew, vnew, out_k, out_v);
  attn_t1<<<BSZ * HEADS, 256, smem_self, stream>>>(qv, out_k, out_v, mask, att, SELFC);
  gemm32<<<EMBED / 128, 256, 0, stream>>>(att, sa_ow, sa_ob, x, x1, EMBED, EMBED, 0);

  // ---- cross-attention ----
  ln_rows<<<BSZ, 256, 0, stream>>>(x1, ln2_g, ln2_b, h);
  gemm32<<<EMBED / 128, 256, 0, stream>>>(h, ca_qw, ca_qb, nullptr, qv, EMBED, EMBED, 1);
  attn_t1<<<BSZ * HEADS, 256, smem_cross, stream>>>(qv, cross_k, cross_v, nullptr, att, CROSSS);
  gemm32<<<EMBED / 128, 256, 0, stream>>>(att, ca_ow, ca_ob, x1, x2, EMBED, EMBED, 0);

  // ---- FFN ----
  ln_rows<<<BSZ, 256, 0, stream>>>(x2, ln3_g, ln3_b, h);
  gemm32<<<FFNDIM / 128, 256, 0, stream>>>(h, fc1_w, fc1_b, nullptr, mid, EMBED, FFNDIM, 2);
  gemm32<<<EMBED / 128, 256, 0, stream>>>(mid, fc2_w, fc2_b, x2, out_x, FFNDIM, EMBED, 0);
}